// GCN_classifier_90443421319565
// MI455X (gfx1250) — compile-verified
//
#include <hip/hip_runtime.h>

// GCN classifier, algebraically collapsed:
//   out = prop(x) @ (W1@W2) + bias*sum(W2) + b2
// prop = D^-1/2 (A + I) D^-1/2.  Memory-bound: ~38 MB HBM traffic total.
// The dense step (s = X @ w) uses V_WMMA_F32_16X16X4_F32 (fp32, exact).

typedef __attribute__((ext_vector_type(2))) float v2f;
typedef __attribute__((ext_vector_type(8))) float v8f;

#define F_DIM 48

// ---- collapse W1@W2 -> w[48];  c = bias[0]*sum(W2) + b2[0] ----
__global__ void prep_w_kernel(const float* __restrict__ W1,
                              const float* __restrict__ bias,
                              const float* __restrict__ W2,
                              const float* __restrict__ b2,
                              float* __restrict__ w,
                              float* __restrict__ cconst,
                              int H) {
  int t = threadIdx.x;
  if (t < F_DIM) {
    float acc = 0.f;
    for (int h = 0; h < H; ++h) acc = fmaf(W1[t * H + h], W2[h], acc);
    w[t] = acc;
  } else if (t == F_DIM) {
    float sw = 0.f;
    for (int h = 0; h < H; ++h) sw += W2[h];
    *cconst = bias[0] * sw + b2[0];
  }
}

// ---- degree: deg[n] = 1 (self loop) + #incoming edges ----
__global__ void init_deg_kernel(float* __restrict__ deg, int N) {
  int n = blockIdx.x * blockDim.x + threadIdx.x;
  if (n < N) deg[n] = 1.0f;
}

__global__ void degree_kernel(const int* __restrict__ col,
                              float* __restrict__ deg, int E) {
  int e = blockIdx.x * blockDim.x + threadIdx.x;
  if (e < E) atomicAdd(&deg[col[e]], 1.0f);
}

// in place: deg -> dinv = rsqrt(max(deg,1))
__global__ void dinv_kernel(float* __restrict__ deg, int N) {
  int n = blockIdx.x * blockDim.x + threadIdx.x;
  if (n < N) deg[n] = rsqrtf(fmaxf(deg[n], 1.0f));
}

// ---- s[n] = dot(x[n,:48], w) via V_WMMA_F32_16X16X4_F32 ----
// One wave handles a 16-node tile. A: 16x4 fp32 per step (lanes 0-15 carry
// K={0,1}, lanes 16-31 carry K={2,3}); B broadcast of w over all 16 columns.
// Chain 12 steps to cover K=48. D column n is identical for all n, so lanes
// 0 and 16 each store 8 results (rows 0-7 / 8-15).
__global__ void s_wmma_kernel(const float* __restrict__ x,
                              const float* __restrict__ w,
                              float* __restrict__ s, int nTiles) {
  int lane = threadIdx.x & 31;
  int tile = blockIdx.x * (blockDim.x >> 5) + (threadIdx.x >> 5);
  if (tile >= nTiles) return;            // wave-uniform: EXEC stays all-1s
  int m = lane & 15;                      // node row within tile
  int halfSel = lane >> 4;                // 0: K pair {0,1}; 1: K pair {2,3}
  const float* xrow = x + (size_t)(tile * 16 + m) * F_DIM;
  v8f c = {0.f, 0.f, 0.f, 0.f, 0.f, 0.f, 0.f, 0.f};
#pragma unroll
  for (int k0 = 0; k0 < F_DIM; k0 += 4) {
    int koff = k0 + 2 * halfSel;
    v2f a = *reinterpret_cast<const v2f*>(xrow + koff);  // 8B aligned
    v2f b = *reinterpret_cast<const v2f*>(w + koff);
    c = __builtin_amdgcn_wmma_f32_16x16x4_f32(
        /*neg_a=*/false, a, /*neg_b=*/false, b,
        /*c_mod=*/(short)0, c, /*reuse_a=*/false, /*reuse_b=*/false);
  }
  if ((lane & 15) == 0) {                 // lanes 0 and 16 hold rows 0-7 / 8-15
    int base = tile * 16 + halfSel * 8;
#pragma unroll
    for (int j = 0; j < 8; ++j) s[base + j] = c[j];
  }
}

// scalar tail for N % 16 nodes (zero for this problem size, kept for safety)
__global__ void s_tail_kernel(const float* __restrict__ x,
                              const float* __restrict__ w,
                              float* __restrict__ s, int start, int N) {
  int n = start + blockIdx.x * blockDim.x + threadIdx.x;
  if (n < N) {
    float acc = 0.f;
    for (int k = 0; k < F_DIM; ++k)
      acc = fmaf(x[(size_t)n * F_DIM + k], w[k], acc);
    s[n] = acc;
  }
}

// ---- out[n] = dinv[n]^2 * s[n] + c   (self-loop term + constant) ----
__global__ void self_out_kernel(const float* __restrict__ dinv,
                                const float* __restrict__ s,
                                const float* __restrict__ cconst,
                                float* __restrict__ out, int N) {
  int n = blockIdx.x * blockDim.x + threadIdx.x;
  if (n < N) {
    float d = dinv[n];
    out[n] = fmaf(d * d, s[n], *cconst);
  }
}

// ---- out[col] += dinv[row]*dinv[col]*s[row] over real edges ----
__global__ void edge_scatter_kernel(const int* __restrict__ row,
                                    const int* __restrict__ col,
                                    const float* __restrict__ dinv,
                                    const float* __restrict__ s,
                                    float* __restrict__ out, int E) {
  int e = blockIdx.x * blockDim.x + threadIdx.x;
  if (e < E) {
    int r = row[e], c = col[e];
    atomicAdd(&out[c], dinv[r] * dinv[c] * s[r]);
  }
}

static inline size_t roundup256(size_t v) { return (v + 255) & ~(size_t)255; }

extern "C" void kernel_launch(void* const* d_in, const int* in_sizes, int n_in,
                              void* d_out, int out_size, void* d_ws, size_t ws_size,
                              hipStream_t stream) {
  const float* x    = (const float*)d_in[0];
  const int*   eidx = (const int*)d_in[1];
  const float* W1   = (const float*)d_in[2];
  const float* bias = (const float*)d_in[3];
  const float* W2   = (const float*)d_in[4];
  const float* b2   = (const float*)d_in[5];
  float* out = (float*)d_out;

  const int N = out_size;            // 100000
  const int E = in_sizes[1] / 2;     // 1600000
  const int H = in_sizes[4];         // 256
  const int* row = eidx;             // sources
  const int* col = eidx + E;         // destinations

  // workspace: deg/dinv (in place), s, w, c
  char* ws = (char*)d_ws;
  float* deg = (float*)ws;                                        // N floats
  float* s   = (float*)(ws + roundup256((size_t)N * 4));          // N floats
  float* w   = (float*)(ws + 2 * roundup256((size_t)N * 4));      // 48 floats
  float* cc  = w + 64;                                            // 1 float
  (void)ws_size; (void)n_in;

  const int T = 256;
  const int nBlkN = (N + T - 1) / T;
  const int nBlkE = (E + T - 1) / T;

  prep_w_kernel<<<1, 64, 0, stream>>>(W1, bias, W2, b2, w, cc, H);
  init_deg_kernel<<<nBlkN, T, 0, stream>>>(deg, N);
  degree_kernel<<<nBlkE, T, 0, stream>>>(col, deg, E);
  dinv_kernel<<<nBlkN, T, 0, stream>>>(deg, N);

  const int nTiles = N / 16;                      // 6250 full 16-node tiles
  const int wavesPerBlk = T / 32;                 // 8 tiles per block
  const int nBlkTile = (nTiles + wavesPerBlk - 1) / wavesPerBlk;
  if (nTiles > 0)
    s_wmma_kernel<<<nBlkTile, T, 0, stream>>>(x, w, s, nTiles);
  const int rem = N - nTiles * 16;
  if (rem > 0)
    s_tail_kernel<<<1, T, 0, stream>>>(x, w, s, nTiles * 16, N);

  self_out_kernel<<<nBlkN, T, 0, stream>>>(deg, s, cc, out, N);
  edge_scatter_kernel<<<nBlkE, T, 0, stream>>>(row, col, deg, s, out, E);
}